// GCNDecoder_30700426232009
// MI455X (gfx1250) — compile-verified
//
#include <hip/hip_runtime.h>

// GCNConv (PyG-style) + dropout + ReLU for MI455X (gfx1250, wave32).
//   out = relu( Â · ((h ⊙ dropout_mask) @ W) + b ),  Â = D^-1/2 (A+I) D^-1/2
// Phases: init -> degree(atomic) -> rsqrt -> WMMA GEMM -> edge scatter(atomic) -> finalize.

#define HDIM 128
#define DROP 0.1f
#define KEEP_INV (1.0f / 0.9f)

typedef float v2f __attribute__((ext_vector_type(2)));
typedef float v8f __attribute__((ext_vector_type(8)));

// ---------------------------------------------------------------- init
// Zero the output accumulator; seed degree with 1.0 (self-loop).
__global__ __launch_bounds__(256) void k_init(float* __restrict__ out,
                                              float* __restrict__ deg, int n) {
    size_t idx = (size_t)blockIdx.x * blockDim.x + threadIdx.x;
    size_t tot = (size_t)n * HDIM;
    if (idx < tot) out[idx] = 0.0f;
    if (idx < (size_t)n) deg[idx] = 1.0f;
}

// ---------------------------------------------------------------- degree
__global__ __launch_bounds__(256) void k_degree(const int* __restrict__ dst,
                                                float* __restrict__ deg, int ne) {
    int e = blockIdx.x * blockDim.x + threadIdx.x;
    if (e < ne) atomicAdd(&deg[dst[e]], 1.0f);
}

// ---------------------------------------------------------------- rsqrt
// deg >= 1 always (self-loop), so no zero guard needed.
__global__ __launch_bounds__(256) void k_rsqrt(float* __restrict__ deg, int n) {
    int i = blockIdx.x * blockDim.x + threadIdx.x;
    if (i < n) deg[i] = rsqrtf(deg[i]);
}

// ---------------------------------------------------------------- GEMM (WMMA f32)
// xw = (h * dropout_mask) @ W.  256 threads = 8 waves; block covers 128 rows.
// Per wave: 16 rows, A tile (16x128 f32) cached in 64 VGPRs in the ISA's
// 16x4 f32 A layout; W staged in LDS; 8 N-tiles x 32 v_wmma_f32_16x16x4_f32.
__global__ __launch_bounds__(256) void k_gemm(const float* __restrict__ h,
                                              const float* __restrict__ u,
                                              const float* __restrict__ W,
                                              float* __restrict__ xw, int n) {
    __shared__ float Wlds[HDIM * HDIM];  // 64 KB, row-major [k][col]

    int tid = threadIdx.x;
    #pragma unroll 4
    for (int i = tid; i < HDIM * HDIM; i += 256) Wlds[i] = W[i];
    __syncthreads();

    const int wave = tid >> 5;
    const int lane = tid & 31;
    const bool hi  = lane >= 16;      // lanes 16..31 hold K = 4s+2, 4s+3
    const int  m0  = blockIdx.x * 128 + wave * 16;
    const int  row = m0 + (lane & 15);
    const int  rld = row < n ? row : (n - 1);   // clamp loads, guard stores

    // ---- load A tile with dropout: a[2s] = K=4s+base, a[2s+1] = K=4s+base+1
    float a[64];
    const float2* h2 = (const float2*)(h + (size_t)rld * HDIM);
    const float2* u2 = (const float2*)(u + (size_t)rld * HDIM);
    #pragma unroll
    for (int s = 0; s < 32; ++s) {
        int k2 = 2 * s + (hi ? 1 : 0);          // float2 index: K = 4s + 2*hi
        float2 xv = h2[k2];
        float2 uv = u2[k2];
        a[2 * s]     = (uv.x >= DROP) ? xv.x * KEEP_INV : 0.0f;
        a[2 * s + 1] = (uv.y >= DROP) ? xv.y * KEEP_INV : 0.0f;
    }

    const int col  = lane & 15;
    const int kofs = hi ? 2 : 0;

    #pragma unroll 1
    for (int nt = 0; nt < 8; ++nt) {
        const int n0 = nt * 16;
        v8f c = {0.f, 0.f, 0.f, 0.f, 0.f, 0.f, 0.f, 0.f};
        #pragma unroll
        for (int s = 0; s < 32; ++s) {
            v2f av;
            av.x = a[2 * s];
            av.y = a[2 * s + 1];
            const int kb = 4 * s + kofs;        // B: lanes 0-15 -> K=4s, hi -> K=4s+2
            v2f bv;
            bv.x = Wlds[kb * HDIM + n0 + col];
            bv.y = Wlds[(kb + 1) * HDIM + n0 + col];
            c = __builtin_amdgcn_wmma_f32_16x16x4_f32(
                    false, av, false, bv, (short)0, c, false, false);
        }
        // C layout: VGPR r -> (M=r, lanes 0-15) / (M=r+8, lanes 16-31), N = lane&15
        #pragma unroll
        for (int r = 0; r < 8; ++r) {
            int orow = m0 + r + (hi ? 8 : 0);
            if (orow < n) xw[(size_t)orow * HDIM + n0 + col] = c[r];
        }
    }
}

// ---------------------------------------------------------------- edge scatter
// One wave per edge; each lane handles 4 columns (float4 gather + 4 f32 atomics).
__global__ __launch_bounds__(256) void k_scatter(const int* __restrict__ src,
                                                 const int* __restrict__ dst,
                                                 const float* __restrict__ dinv,
                                                 const float* __restrict__ xw,
                                                 float* __restrict__ out, int ne) {
    long long gid = (long long)blockIdx.x * blockDim.x + threadIdx.x;
    int e = (int)(gid >> 5);
    if (e >= ne) return;
    int c0 = ((int)gid & 31) * 4;
    int s = src[e];
    int d = dst[e];
    float nrm = dinv[s] * dinv[d];
    const float4 v = *(const float4*)(xw + (size_t)s * HDIM + c0);
    float* od = out + (size_t)d * HDIM + c0;
    atomicAdd(od + 0, v.x * nrm);
    atomicAdd(od + 1, v.y * nrm);
    atomicAdd(od + 2, v.z * nrm);
    atomicAdd(od + 3, v.w * nrm);
}

// ---------------------------------------------------------------- finalize
// out = relu(out + xw * dinv^2 (self-loop) + bias)
__global__ __launch_bounds__(256) void k_final(float* __restrict__ out,
                                               const float* __restrict__ xw,
                                               const float* __restrict__ dinv,
                                               const float* __restrict__ b, int n) {
    size_t idx = (size_t)blockIdx.x * blockDim.x + threadIdx.x;
    if (idx >= (size_t)n * HDIM) return;
    int i = (int)(idx >> 7);
    int j = (int)(idx & 127);
    float di = dinv[i];
    float v = out[idx] + xw[idx] * di * di + b[j];
    out[idx] = fmaxf(v, 0.0f);
}

// ---------------------------------------------------------------- launcher
extern "C" void kernel_launch(void* const* d_in, const int* in_sizes, int n_in,
                              void* d_out, int out_size, void* d_ws, size_t ws_size,
                              hipStream_t stream) {
    const float* h     = (const float*)d_in[0];
    const int*   edges = (const int*)d_in[1];   // [2, E] flat; row0=src, row1=dst
    const float* u     = (const float*)d_in[2];
    const float* W     = (const float*)d_in[3];
    const float* b     = (const float*)d_in[4];

    const int N = in_sizes[0] / HDIM;
    const int E = in_sizes[1] / 2;
    const int* src = edges;
    const int* dst = edges + E;

    float* out = (float*)d_out;
    float* deg = (float*)d_ws;                           // N floats (deg -> dinv)
    size_t degPad = ((size_t)N + 255) & ~(size_t)255;    // keep xw 1KB-aligned
    float* xw  = deg + degPad;                           // N*128 floats

    const int T = 256;
    const size_t tot = (size_t)N * HDIM;

    k_init  <<<(int)((tot + T - 1) / T), T, 0, stream>>>(out, deg, N);
    k_degree<<<(E + T - 1) / T,          T, 0, stream>>>(dst, deg, E);
    k_rsqrt <<<(N + T - 1) / T,          T, 0, stream>>>(deg, N);
    k_gemm  <<<(N + 127) / 128,          T, 0, stream>>>(h, u, W, xw, N);
    long long sthr = (long long)E * 32;
    k_scatter<<<(int)((sthr + T - 1) / T), T, 0, stream>>>(src, dst, deg, xw, out, E);
    k_final <<<(int)((tot + T - 1) / T), T, 0, stream>>>(out, xw, deg, b, N);
}